// FractalDimensionLayer_50483045598001
// MI455X (gfx1250) — compile-verified
//
#include <hip/hip_runtime.h>
#include <hip/hip_bf16.h>

#define T_STEPS 131072
#define FEAT 8
#define HIDDEN 256
#define MAX_K 10
#define BUF 50

typedef __attribute__((ext_vector_type(2))) float v2f;
typedef __attribute__((ext_vector_type(8))) float v8f;

// ---------------------------------------------------------------------------
// Kernel 1: Higuchi fractal dimension per timestep -> hfd[T] (f32 workspace).
// One thread per t; window staged in LDS (stride 51 -> no bank conflicts).
// Compute is trivial (~455 abs-diffs/thread); kernel is latency-bound but tiny
// next to the 134 MB output stage, so simplicity wins.
// ---------------------------------------------------------------------------
__global__ __launch_bounds__(256) void hfd_kernel(const float* __restrict__ x,
                                                  float* __restrict__ hfd_out) {
    __shared__ float Wsh[256][BUF + 1];
    const int t = blockIdx.x * 256 + threadIdx.x;
    float* W = Wsh[threadIdx.x];

    const int n = min(t + 1, BUF);       // buffer length at step t
    const int nm1 = n - 1;
    const float nm1f = (float)nm1;
    const int start = max(0, t + 1 - BUF);

    for (int i = 0; i < n; ++i)
        W[i] = x[(size_t)(start + i) * FEAT + 3];   // close price column

    // log(k) constants; fully-unrolled k loop folds these to immediates.
    const float LOGK[MAX_K + 1] = {0.f, 0.f, 0.69314718f, 1.09861229f,
                                   1.38629436f, 1.60943791f, 1.79175947f,
                                   1.94591015f, 2.07944154f, 2.19722458f,
                                   2.30258509f};

    float c = 0.f, sx = 0.f, sy = 0.f, sxy = 0.f, sxx = 0.f;
#pragma unroll
    for (int k = 1; k <= MAX_K; ++k) {
        float acc = 0.f;
        for (int m = 0; m < k; ++m) {
            float S = 0.f;
            for (int q = m; q + k <= nm1; q += k)
                S += fabsf(W[q + k] - W[q]);
            const int length = (nm1 >= m) ? ((nm1 - m) / k + 1) : 0;
            if (length >= 2)
                acc += S * nm1f / ((float)length * (float)k);
        }
        const float Lk = acc / (float)k;
        if (Lk > 0.f) {
            const float lk = LOGK[k];
            const float ly = logf(Lk);
            c += 1.f; sx += lk; sy += ly; sxy += ly * lk; sxx += lk * lk;
        }
    }
    const float denom = c * sxx - sx * sx;
    const float safe  = (denom != 0.f) ? denom : 1.f;
    const float slope = (c * sxy - sx * sy) / safe;
    const float h = ((c > 1.f) && (n >= MAX_K + 1)) ? -slope : 0.f;
    hfd_out[t] = h;
}

// ---------------------------------------------------------------------------
// Kernel 2: out[t,h] = relu(hfd[t] * w[h] + b[h]) via V_WMMA_F32_16X16X4_F32.
// Rank-1 update expressed as a K=4 f32 WMMA with K=1..3 zero-padded:
//   A (16x4): col0 = hfd[t0..t0+15], cols 1..3 = 0
//   B (4x16): row0 = w[n0..n0+15],  rows 1..3 = 0
//   C (16x16): bias broadcast along rows
// One wave per 16x16 tile; 8 waves / 256-thread block; EXEC all-ones at WMMA.
// Kernel is store-BW bound (134 MB) -> WMMA cost is hidden.
// ---------------------------------------------------------------------------
__global__ __launch_bounds__(256) void expand_wmma_kernel(const float* __restrict__ hfd,
                                                          const float* __restrict__ w,
                                                          const float* __restrict__ b,
                                                          float* __restrict__ out) {
    const int lane = threadIdx.x & 31;
    const int wave = threadIdx.x >> 5;
    const int tile = blockIdx.x * 8 + wave;        // 131072 tiles total
    const int colTile = tile & (HIDDEN / 16 - 1);  // 16 col tiles
    const int rowTile = tile >> 4;                 // 8192 row tiles
    const int n0 = colTile * 16;
    const int t0 = rowTile * 16;
    const int l15 = lane & 15;

    // Unconditional loads (clamped index) + v_cndmask -> no EXEC divergence.
    const float av = hfd[t0 + l15];      // A col K=0, M = lane (lanes 0-15)
    const float bv = w[n0 + l15];        // B row K=0, N = lane (lanes 0-15)
    const float bias = b[n0 + l15];      // every C/D VGPR: N = lane%16

    v2f A; A.x = (lane < 16) ? av : 0.f; A.y = 0.f;  // lanes 16-31 hold K=2/3 -> 0
    v2f B; B.x = (lane < 16) ? bv : 0.f; B.y = 0.f;

    v8f C;
#pragma unroll
    for (int r = 0; r < 8; ++r) C[r] = bias;

    // 8 args: (neg_a, A, neg_b, B, c_mod, C, reuse_a, reuse_b)
    v8f D = __builtin_amdgcn_wmma_f32_16x16x4_f32(
        false, A, false, B, (short)0, C, false, false);

    // D layout: VGPR r -> M=r (lanes 0-15) / M=r+8 (lanes 16-31), N=lane%16.
    const int rowBase = t0 + ((lane >> 4) << 3);
    float* outp = out + (size_t)rowBase * HIDDEN + n0 + l15;
#pragma unroll
    for (int r = 0; r < 8; ++r) {
        float v = D[r];
        v = (v > 0.f) ? v : 0.f;                    // relu
        outp[(size_t)r * HIDDEN] = v;
    }
}

extern "C" void kernel_launch(void* const* d_in, const int* in_sizes, int n_in,
                              void* d_out, int out_size, void* d_ws, size_t ws_size,
                              hipStream_t stream) {
    const float* x     = (const float*)d_in[0];   // [T, 8] f32
    const float* w_lin = (const float*)d_in[1];   // [HIDDEN, 1] f32
    const float* b_lin = (const float*)d_in[2];   // [HIDDEN] f32
    float* out = (float*)d_out;                   // [T, HIDDEN] f32
    float* hfd = (float*)d_ws;                    // T floats of scratch (512 KB)

    hfd_kernel<<<T_STEPS / 256, 256, 0, stream>>>(x, hfd);

    const int tiles = (T_STEPS / 16) * (HIDDEN / 16);   // 131072
    expand_wmma_kernel<<<tiles / 8, 256, 0, stream>>>(hfd, w_lin, b_lin, out);
}